// SoundSourceLocalize_48481590837460
// MI455X (gfx1250) — compile-verified
//
#include <hip/hip_runtime.h>

typedef __attribute__((ext_vector_type(2))) float v2f;
typedef __attribute__((ext_vector_type(8))) float v8f;

#define SRATE    16000.0f
#define NFFT     512
#define HOP      256
#define NBINS    257          // rfft bins kept (all of them: 8000 Hz <= 8575 Hz)
#define NPADC    544          // padded spec cols: [0,257) real, [272,529) imag
#define IMOFF    272
#define TFRAMES  251
#define BATCH    64
#define NSIG     128          // 64 batches * 2 channels (bc = 2*b + c)
#define NSAMP    64000
#define MROWS    (NSIG * TFRAMES)   // 32128 = 2008 * 16
#define MTILES   (MROWS / 16)       // 2008
#define NTILES   (NPADC / 16)       // 34
#define NANG     1801
#define EPSF     1e-6f
#define PREC     0.97f
#define INV16    (1.0f / 32768.0f)
#define PI_F     3.14159265358979323846f

// ---------------------------------------------------------------- means
__global__ void k_mean(const float* __restrict__ L, const float* __restrict__ R,
                       float* __restrict__ means) {
    __shared__ float red[256];
    int bc = blockIdx.x;
    const float* x = (bc & 1) ? (R + (size_t)(bc >> 1) * NSAMP)
                              : (L + (size_t)(bc >> 1) * NSAMP);
    float s = 0.0f;
    for (int i = threadIdx.x; i < NSAMP; i += 256) s += x[i];
    red[threadIdx.x] = s; __syncthreads();
    for (int o = 128; o > 0; o >>= 1) {
        if (threadIdx.x < o) red[threadIdx.x] += red[threadIdx.x + o];
        __syncthreads();
    }
    if (threadIdx.x == 0) means[bc] = red[0] * INV16 * (1.0f / (float)NSAMP);
}

// ---------------------------------------------------------------- temporal weights
__global__ void k_wn(float* __restrict__ wn) {
    __shared__ float red[256];
    __shared__ float denom;
    int t = threadIdx.x;
    float tw = (t < TFRAMES) ? expf(-2.0f + 2.0f * (float)t / 4095.0f) : 0.0f;
    red[t] = tw; __syncthreads();
    for (int o = 128; o > 0; o >>= 1) {
        if (t < o) red[t] += red[t + o];
        __syncthreads();
    }
    if (t == 0) denom = red[0] + EPSF;
    __syncthreads();
    if (t < TFRAMES) wn[t] = tw / denom;
}

// ---------------------------------------------------------------- DFT basis (window folded in)
__global__ void k_basis(float* __restrict__ B) {
    int gid = blockIdx.x * 256 + threadIdx.x;       // < 512*544
    int n = gid / NPADC, j = gid - n * NPADC;
    float win = 0.5f - 0.5f * cosf((2.0f * PI_F / NFFT) * (float)n);
    float v = 0.0f;
    int k = -1; bool isIm = false;
    if (j < NBINS)                         { k = j; }
    else if (j >= IMOFF && j < IMOFF + NBINS) { k = j - IMOFF; isIm = true; }
    if (k >= 0) {
        int r = (n * k) & (NFFT - 1);               // exact phase reduction
        float th = (2.0f * PI_F / NFFT) * (float)r;
        v = isIm ? (-win * sinf(th)) : (win * cosf(th));
    }
    B[gid] = v;
}

// ---------------------------------------------------------------- steering tables
__global__ void k_steer(float* __restrict__ SRm, float* __restrict__ SIm) {
    int gid = blockIdx.x * 256 + threadIdx.x;
    if (gid >= NANG * NBINS) return;
    int a = gid / NBINS, f = gid - a * NBINS;
    float rad = (float)a * (180.0f / 1800.0f) * (PI_F / 180.0f);
    float tau = -(0.02f * cosf(rad)) / 343.0f;
    float fz  = (float)f * (SRATE / (float)NFFT);
    float ph  = 2.0f * PI_F * tau * fz;
    SRm[gid] = cosf(ph);
    SIm[gid] = sinf(ph);
}

// ---------------------------------------------------------------- frequency weights
__global__ void k_cw(float* __restrict__ cw) {
    __shared__ float raw[NBINS];
    __shared__ float s_sum;
    for (int f = threadIdx.x; f < NBINS; f += 256) {
        float fz = (float)f * (SRATE / (float)NFFT);
        float sres = 0.02f * (fz + EPSF) / 343.0f;
        float sw = 1.0f / (1.0f + expf(-4.0f * (sres - 0.1f)));
        float center = SRATE / 6.0f, sigma = SRATE / 4.0f;
        float d = fz - center;
        float fw = 1.0f - expf(-(d * d) / (2.0f * sigma * sigma));
        raw[f] = sw * fw;
    }
    __syncthreads();
    if (threadIdx.x == 0) {
        float s = 0.0f;
        for (int f = 0; f < NBINS; ++f) s += raw[f];
        s_sum = s + EPSF;
    }
    __syncthreads();
    for (int f = threadIdx.x; f < NBINS; f += 256) cw[f] = raw[f] / s_sum;
}

// ---------------------------------------------------------------- framing (mean-sub + pre-emphasis + window)
__global__ void k_frames(const float* __restrict__ L, const float* __restrict__ R,
                         const float* __restrict__ means, float* __restrict__ Fr) {
    int gid = blockIdx.x * 256 + threadIdx.x;       // < MROWS*512
    int m = gid >> 9, n = gid & 511;
    int bc = m / TFRAMES, t = m - bc * TFRAMES;
    const float* x = (bc & 1) ? (R + (size_t)(bc >> 1) * NSAMP)
                              : (L + (size_t)(bc >> 1) * NSAMP);
    float mean = means[bc];
    int idx = t * HOP + n - (NFFT / 2);
    float v = 0.0f;
    if (idx >= 0 && idx < NSAMP) {
        float s = (idx == 0) ? (x[0] * INV16 - mean)
                             : ((x[idx] - PREC * x[idx - 1]) * INV16 - (1.0f - PREC) * mean);
        float win = 0.5f - 0.5f * cosf((2.0f * PI_F / NFFT) * (float)n);
        v = win * s;
    }
    Fr[gid] = v;
}

// ---------------------------------------------------------------- STFT as f32 WMMA GEMM
// C(32128 x 544) = A(32128 x 512) * B(512 x 544), one 16x16 tile per wave32.
__global__ void __launch_bounds__(256) k_gemm(const float* __restrict__ A,
                                              const float* __restrict__ B,
                                              float* __restrict__ C) {
    int wave = (blockIdx.x << 3) + (threadIdx.x >> 5);
    int lane = threadIdx.x & 31;
    int half = lane >> 4;           // 0: lanes 0-15, 1: lanes 16-31
    int l16  = lane & 15;
    int mT = wave % MTILES;
    int nT = wave / MTILES;
    const float* Ar = A + (size_t)(mT * 16 + l16) * NFFT;   // A row for M = l16
    const float* Bc = B + nT * 16 + l16;                    // B col = nT*16 + l16
    v8f acc = {0.0f, 0.0f, 0.0f, 0.0f, 0.0f, 0.0f, 0.0f, 0.0f};
    #pragma unroll 8
    for (int kb = 0; kb < NFFT; kb += 4) {
        int ka = kb + (half << 1);                          // K = 2*half (+1)
        v2f av; av.x = Ar[ka];                 av.y = Ar[ka + 1];
        v2f bv; bv.x = Bc[(size_t)ka * NPADC]; bv.y = Bc[(size_t)(ka + 1) * NPADC];
        acc = __builtin_amdgcn_wmma_f32_16x16x4_f32(
                  false, av, false, bv, (short)0, acc, false, false);
    }
    size_t colBase = (size_t)nT * 16 + l16;
    #pragma unroll
    for (int j = 0; j < 8; ++j) {
        int row = mT * 16 + j + (half << 3);                // C layout: VGPR j -> M=j(+8)
        C[(size_t)row * NPADC + colBase] = acc[j];
    }
}

// ---------------------------------------------------------------- covariance + MVDR prep per (b,f)
__global__ void k_cov(const float* __restrict__ spec, const float* __restrict__ wn,
                      float* __restrict__ cDen, float* __restrict__ cIRr,
                      float* __restrict__ cIRi, float* __restrict__ cCm,
                      float* __restrict__ cSm,  float* __restrict__ cCoh) {
    int i = blockIdx.x * 256 + threadIdx.x;
    if (i >= BATCH * NBINS) return;
    int b = i / NBINS, f = i - b * NBINS;
    const float* sL = spec + (size_t)(2 * b)     * TFRAMES * NPADC;
    const float* sR = spec + (size_t)(2 * b + 1) * TFRAMES * NPADC;
    float Rll = 0.f, Rrr = 0.f, Crr = 0.f, Cii = 0.f, g2 = 0.f;
    for (int t = 0; t < TFRAMES; ++t) {
        const float* pL = sL + (size_t)t * NPADC;
        const float* pR = sR + (size_t)t * NPADC;
        float rL = pL[f], iL = pL[IMOFF + f];
        float rR = pR[f], iR = pR[IMOFF + f];
        float powL = rL * rL + iL * iL;
        float powR = rR * rR + iR * iR;
        float xr = rL * rR + iL * iR;
        float xi = iL * rR - rL * iR;
        float w = wn[t];
        Rll += powL * w; Rrr += powR * w; Crr += xr * w; Cii += xi * w;
        float gf = (xr * xr + xi * xi) / (powL * powR + EPSF);
        g2 += fminf(fmaxf(gf, 0.0f), 1.0f) * w;
    }
    float prod = Rll * Rrr;
    float cmag = sqrtf(Crr * Crr + Cii * Cii);
    float cmax = 0.999f * sqrtf(fmaxf(prod, 0.0f));
    float cs = fminf(cmax / (cmag + EPSF), 1.0f);
    Crr *= cs; Cii *= cs;
    float trace = (Rll + Rrr) * 0.5f;
    float det = prod - (Crr * Crr + Cii * Cii);
    float cond = trace / (fabsf(det) + EPSF);
    float loading = fminf(fmaxf(0.001f * cond, 1e-6f), 0.01f);
    det += loading * trace;
    float invDet = 1.0f / (det + EPSF);
    cDen[i] = (Rrr + Rll) * invDet;      // iR_LL + iR_RR
    cIRr[i] = -Crr * invDet;
    cIRi[i] = -Cii * invDet;
    float sn = rsqrtf(Crr * Crr + Cii * Cii + EPSF);
    cCm[i] = Crr * sn;
    cSm[i] = Cii * sn;
    cCoh[i] = sqrtf(fminf(fmaxf(g2, 0.0f), 1.0f));
}

// ---------------------------------------------------------------- mpg evaluation (shared by both passes)
__device__ __forceinline__ float mpgv(float den, float irr, float iri, float cm,
                                      float sm, float ch, float sr, float si) {
    float quad = irr * sr + iri * si;
    float d = den + quad + quad;
    float mvdr = 1.0f / (d + EPSF);
    float cosd = cm * sr + sm * si;
    float gating = 0.2f + 0.8f * (0.5f * (1.0f + cosd) * ch);
    return mvdr * gating + EPSF;
}

// ---------------------------------------------------------------- pass A: softmax stats over angles per (b,f)
__global__ void k_passA(const float* __restrict__ SRm, const float* __restrict__ SIm,
                        const float* __restrict__ cDen, const float* __restrict__ cIRr,
                        const float* __restrict__ cIRi, const float* __restrict__ cCm,
                        const float* __restrict__ cSm,  const float* __restrict__ cCoh,
                        float* __restrict__ maxv, float* __restrict__ sumexp) {
    __shared__ float red[256];
    int bf = blockIdx.x;                 // b*257 + f
    int f = bf % NBINS;
    float den = cDen[bf], irr = cIRr[bf], iri = cIRi[bf];
    float cm = cCm[bf], sm = cSm[bf], ch = cCoh[bf];
    float vals[8];
    float mx = -3.0e38f;
    #pragma unroll
    for (int i = 0; i < 8; ++i) {
        int a = threadIdx.x + (i << 8);
        float v = -3.0e38f;
        if (a < NANG) {
            int af = a * NBINS + f;
            v = mpgv(den, irr, iri, cm, sm, ch, SRm[af], SIm[af]);
        }
        vals[i] = v;
        mx = fmaxf(mx, v);
    }
    red[threadIdx.x] = mx; __syncthreads();
    for (int o = 128; o > 0; o >>= 1) {
        if (threadIdx.x < o) red[threadIdx.x] = fmaxf(red[threadIdx.x], red[threadIdx.x + o]);
        __syncthreads();
    }
    float M = red[0]; __syncthreads();
    float se = 0.0f;
    #pragma unroll
    for (int i = 0; i < 8; ++i) {
        int a = threadIdx.x + (i << 8);
        if (a < NANG) se += expf(vals[i] - M);
    }
    red[threadIdx.x] = se; __syncthreads();
    for (int o = 128; o > 0; o >>= 1) {
        if (threadIdx.x < o) red[threadIdx.x] += red[threadIdx.x + o];
        __syncthreads();
    }
    if (threadIdx.x == 0) { maxv[bf] = M; sumexp[bf] = red[0]; }
}

// ---------------------------------------------------------------- pass B: fused score per (b,a)
__global__ void k_passB(const float* __restrict__ SRm, const float* __restrict__ SIm,
                        const float* __restrict__ cDen, const float* __restrict__ cIRr,
                        const float* __restrict__ cIRi, const float* __restrict__ cCm,
                        const float* __restrict__ cSm,  const float* __restrict__ cCoh,
                        const float* __restrict__ maxv, const float* __restrict__ sumexp,
                        const float* __restrict__ cw, float* __restrict__ score) {
    int idx = blockIdx.x * 256 + threadIdx.x;
    if (idx >= BATCH * NANG) return;
    int b = idx / NANG, a = idx - b * NANG;
    float wlog = 0.f, ar = 0.f, pk = 0.f;
    for (int f = 0; f < NBINS; ++f) {
        int bf = b * NBINS + f, af = a * NBINS + f;
        float v = mpgv(cDen[bf], cIRr[bf], cIRi[bf], cCm[bf], cSm[bf], cCoh[bf],
                       SRm[af], SIm[af]);
        float w = cw[f];
        wlog += logf(v) * w;
        ar   += v * w;
        pk   += v * (expf(v - maxv[bf]) / sumexp[bf]) * w;
    }
    score[idx] = 0.4f * expf(wlog) + 0.4f * ar + 0.2f * pk;
}

// ---------------------------------------------------------------- argmax + parabolic refine per batch
__global__ void k_final(const float* __restrict__ score, float* __restrict__ out) {
    __shared__ float sv[256];
    __shared__ int   si[256];
    int b = blockIdx.x;
    const float* s = score + (size_t)b * NANG;
    float bv = -3.0e38f; int bi = 0;
    for (int a = threadIdx.x; a < NANG; a += 256) {
        float v = s[a];
        if (v > bv) { bv = v; bi = a; }
    }
    sv[threadIdx.x] = bv; si[threadIdx.x] = bi; __syncthreads();
    for (int o = 128; o > 0; o >>= 1) {
        if (threadIdx.x < o) {
            float ov = sv[threadIdx.x + o]; int oi = si[threadIdx.x + o];
            if (ov > sv[threadIdx.x] || (ov == sv[threadIdx.x] && oi < si[threadIdx.x])) {
                sv[threadIdx.x] = ov; si[threadIdx.x] = oi;
            }
        }
        __syncthreads();
    }
    if (threadIdx.x == 0) {
        int i1 = si[0];
        int i0 = max(i1 - 1, 0), i2 = min(i1 + 1, NANG - 1);
        float y0 = s[i0], y1 = s[i1], y2 = s[i2];
        float dn = y0 - 2.0f * y1 + y2;
        float off = (fabsf(dn) > 1e-12f) ? 0.5f * (y0 - y2) / dn : 0.0f;
        off = fminf(fmaxf(off, -1.0f), 1.0f);
        float ref = (float)i1 * 0.1f + off;
        out[b] = fminf(fmaxf(ref, 0.0f), 180.0f);
    }
}

// ---------------------------------------------------------------- launch
extern "C" void kernel_launch(void* const* d_in, const int* in_sizes, int n_in,
                              void* d_out, int out_size, void* d_ws, size_t ws_size,
                              hipStream_t stream) {
    const float* L = (const float*)d_in[0];
    const float* R = (const float*)d_in[1];
    float* out = (float*)d_out;

    char* w = (char*)d_ws;
    size_t off = 0;
    auto carve = [&](size_t nElems) -> float* {
        float* p = (float*)(w + off);
        off = (off + nElems * sizeof(float) + 255) & ~(size_t)255;
        return p;
    };
    float* means  = carve(NSIG);
    float* wn     = carve(TFRAMES);
    float* Bmat   = carve((size_t)NFFT * NPADC);
    float* SRm    = carve((size_t)NANG * NBINS);
    float* SIm    = carve((size_t)NANG * NBINS);
    float* cw     = carve(NBINS);
    float* cDen   = carve((size_t)BATCH * NBINS);
    float* cIRr   = carve((size_t)BATCH * NBINS);
    float* cIRi   = carve((size_t)BATCH * NBINS);
    float* cCm    = carve((size_t)BATCH * NBINS);
    float* cSm    = carve((size_t)BATCH * NBINS);
    float* cCoh   = carve((size_t)BATCH * NBINS);
    float* maxv   = carve((size_t)BATCH * NBINS);
    float* sumexp = carve((size_t)BATCH * NBINS);
    float* score  = carve((size_t)BATCH * NANG);
    float* frames = carve((size_t)MROWS * NFFT);    // 65.8 MB
    float* spec   = carve((size_t)MROWS * NPADC);   // 69.9 MB

    k_mean  <<<NSIG, 256, 0, stream>>>(L, R, means);
    k_wn    <<<1, 256, 0, stream>>>(wn);
    k_basis <<<(NFFT * NPADC) / 256, 256, 0, stream>>>(Bmat);
    k_steer <<<(NANG * NBINS + 255) / 256, 256, 0, stream>>>(SRm, SIm);
    k_cw    <<<1, 256, 0, stream>>>(cw);
    k_frames<<<(MROWS * NFFT) / 256, 256, 0, stream>>>(L, R, means, frames);
    k_gemm  <<<(MTILES * NTILES) / 8, 256, 0, stream>>>(frames, Bmat, spec);
    k_cov   <<<(BATCH * NBINS + 255) / 256, 256, 0, stream>>>(spec, wn, cDen, cIRr, cIRi, cCm, cSm, cCoh);
    k_passA <<<BATCH * NBINS, 256, 0, stream>>>(SRm, SIm, cDen, cIRr, cIRi, cCm, cSm, cCoh, maxv, sumexp);
    k_passB <<<(BATCH * NANG + 255) / 256, 256, 0, stream>>>(SRm, SIm, cDen, cIRr, cIRi, cCm, cSm, cCoh, maxv, sumexp, cw, score);
    k_final <<<BATCH, 256, 0, stream>>>(score, out);
}